// OTContrastiveLoss_7980049236653
// MI455X (gfx1250) — compile-verified
//
#include <hip/hip_runtime.h>
#include <hip/hip_bf16.h>
#include <math.h>

typedef __attribute__((ext_vector_type(2))) float v2f;
typedef __attribute__((ext_vector_type(8))) float v8f;

#define C_DIM 256
#define M_Q   1024
#define N_T   8192
#define LDN   8192
#define OT_ITERS 1000

// workspace layout (float offsets)
#define OFF_M   ((size_t)0)                 // 1024*8192 cost matrix
#define OFF_NRM ((size_t)8388608)           // 9216 row norms (qq: first 1024, tt: rest)
#define OFF_LOG (OFF_NRM + 9216)            // 9216 log(att)  (log_mu first 1024, log_nu rest)
#define OFF_U   (OFF_LOG + 9216)            // 1024
#define OFF_V   (OFF_U + 1024)              // 8192
#define OFF_RD  (OFF_V + 8192)              // 1024 per-row distances

// ---------------------------------------------------------------- prep: norms, log(att), zero u/v
__global__ void ot_prep(const float* __restrict__ x, const float* __restrict__ att,
                        float* __restrict__ ws) {
  int p = blockIdx.x * blockDim.x + threadIdx.x;       // 0..9215
  if (p >= 9216) return;
  int img = p >> 10, pix = p & 1023;
  const float* base = x + (size_t)img * (C_DIM * M_Q) + pix;
  float s = 0.f;
  for (int c = 0; c < C_DIM; ++c) {
    float vv = base[(size_t)c * M_Q];
    s += vv * vv;
  }
  ws[OFF_NRM + p] = s;
  ws[OFF_LOG + p] = __logf(att[p]);
  if (p < 1024) ws[OFF_U + p] = 0.f;
  if (p < 8192) ws[OFF_V + p] = 0.f;
}

// ---------------------------------------------------------------- WMMA f32 GEMM: M = qq + tt - 2 Q T^T
// one wave -> 16x64 output tile (4 accumulators of 16x16), K blocked by 4
__global__ void __launch_bounds__(256) ot_gemm(const float* __restrict__ x, float* __restrict__ ws) {
  const int lane = threadIdx.x & 31;
  const int wave = blockIdx.x * 8 + (threadIdx.x >> 5);
  const int tileM = wave >> 7;      // 64 row tiles
  const int tileN = wave & 127;     // 128 col tiles (64 cols each)
  const int rowbase = tileM << 4;
  const int colbase = tileN << 6;
  const int half = lane >> 4;       // K-half select per ISA layout
  const int mrow = lane & 15;

  v8f acc[4];
#pragma unroll
  for (int t = 0; t < 4; ++t)
#pragma unroll
    for (int r = 0; r < 8; ++r) acc[t][r] = 0.f;

  // Q[i][c] = x[c*1024 + i]; T[j][c] = x[(1+j/1024)*262144 + c*1024 + (j%1024)]
  const size_t aoff = (size_t)(rowbase + mrow);
  size_t boff[4];
#pragma unroll
  for (int t = 0; t < 4; ++t) {
    int j = colbase + (t << 4) + mrow;
    boff[t] = (size_t)(1 + (j >> 10)) * (C_DIM * M_Q) + (size_t)(j & 1023);
  }

  for (int k = 0; k < C_DIM; k += 4) {
    const size_t krow = (size_t)(k + (half << 1)) * M_Q;
    v2f a;
    a.x = x[aoff + krow];
    a.y = x[aoff + krow + M_Q];
#pragma unroll
    for (int t = 0; t < 4; ++t) {
      v2f b;
      b.x = x[boff[t] + krow];
      b.y = x[boff[t] + krow + M_Q];
      acc[t] = __builtin_amdgcn_wmma_f32_16x16x4_f32(false, a, false, b,
                                                     (short)0, acc[t], false, false);
    }
  }

  const float* nrm = ws + OFF_NRM;
#pragma unroll
  for (int r = 0; r < 8; ++r) {
    int i = rowbase + r + (half << 3);
    float qq = nrm[i];
#pragma unroll
    for (int t = 0; t < 4; ++t) {
      int j = colbase + (t << 4) + mrow;
      ws[OFF_M + (size_t)i * LDN + j] = qq + nrm[1024 + j] - 2.f * acc[t][r];
    }
  }
}

// ---------------------------------------------------------------- online logsumexp merge
__device__ __forceinline__ void lse_merge(float& m1, float& s1, float m2, float s2) {
  float mm = fmaxf(m1, m2);
  s1 = s1 * __expf(m1 - mm) + s2 * __expf(m2 - mm);
  m1 = mm;
}

// u[i] = log_mu[i] - logsumexp_j(M[i,j] + v[j]); one block per row, coalesced
__global__ void __launch_bounds__(256) ot_u(const float* __restrict__ Mm, const float* __restrict__ v,
                                            const float* __restrict__ loga, float* __restrict__ u) {
  const int i = blockIdx.x, tid = threadIdx.x;
  const float* row = Mm + (size_t)i * LDN;
  float mx = -INFINITY, s = 0.f;
  for (int j = tid; j < N_T; j += 256) {
    float xv = row[j] + v[j];
    if (xv > mx) { s = s * __expf(mx - xv) + 1.f; mx = xv; }
    else         { s += __expf(xv - mx); }
  }
  __shared__ float sm[256], ss[256];
  sm[tid] = mx; ss[tid] = s;
  __syncthreads();
  for (int off = 128; off > 0; off >>= 1) {
    if (tid < off) {
      float m1 = sm[tid], s1 = ss[tid];
      lse_merge(m1, s1, sm[tid + off], ss[tid + off]);
      sm[tid] = m1; ss[tid] = s1;
    }
    __syncthreads();
  }
  if (tid == 0) u[i] = loga[i] - (sm[0] + __logf(ss[0]));
}

// v[j] = log_nu[j] - logsumexp_i(M[i,j] + u[i]); block = 64 cols x 4 row-chunks
__global__ void __launch_bounds__(256) ot_v(const float* __restrict__ Mm, const float* __restrict__ u,
                                            const float* __restrict__ loga, float* __restrict__ v) {
  __shared__ float us[1024];
  const int tid = threadIdx.x;
  for (int i = tid; i < 1024; i += 256) us[i] = u[i];
  __syncthreads();
  const int c = tid & 63, rc = tid >> 6;
  const int j = blockIdx.x * 64 + c;
  float mx = -INFINITY, s = 0.f;
  for (int i = rc * 256; i < rc * 256 + 256; ++i) {
    float xv = Mm[(size_t)i * LDN + j] + us[i];
    if (xv > mx) { s = s * __expf(mx - xv) + 1.f; mx = xv; }
    else         { s += __expf(xv - mx); }
  }
  __shared__ float sm[256], ss[256];
  sm[tid] = mx; ss[tid] = s;
  __syncthreads();
  if (rc == 0) {
    float m1 = sm[c], s1 = ss[c];
    lse_merge(m1, s1, sm[64 + c],  ss[64 + c]);
    lse_merge(m1, s1, sm[128 + c], ss[128 + c]);
    lse_merge(m1, s1, sm[192 + c], ss[192 + c]);
    v[j] = loga[1024 + j] - (m1 + __logf(s1));
  }
}

// ---------------------------------------------------------------- row distance: || mu[i]*Q[i] - P[i,:] @ T ||
__global__ void __launch_bounds__(256) ot_pt(const float* __restrict__ x, const float* __restrict__ att,
                                             const float* __restrict__ Mm, const float* __restrict__ u,
                                             const float* __restrict__ v, float* __restrict__ rowd) {
  const int i = blockIdx.x, c = threadIdx.x;
  __shared__ float ps[256];
  const float ui = u[i];
  const float* Mrow = Mm + (size_t)i * LDN;
  float acc = 0.f;
  for (int jb = 0; jb < N_T; jb += 256) {
    int j = jb + c;
    ps[c] = __expf(Mrow[j] + ui + v[j]);
    __syncthreads();
    const int img = 1 + (jb >> 10);      // a 256-chunk never crosses an image boundary
    const float* Tc = x + (size_t)img * (C_DIM * M_Q) + (size_t)c * M_Q + (jb & 1023);
#pragma unroll 8
    for (int jj = 0; jj < 256; ++jj) acc += ps[jj] * Tc[jj];
    __syncthreads();
  }
  float q = x[(size_t)c * M_Q + i];      // Q[i][c]
  float diff = att[i] * q - acc;         // mu[i] = att[0 .. 1023]
  __shared__ float red[256];
  red[c] = diff * diff;
  __syncthreads();
  for (int off = 128; off > 0; off >>= 1) {
    if (c < off) red[c] += red[c + off];
    __syncthreads();
  }
  if (c == 0) rowd[i] = sqrtf(red[0]);
}

// ---------------------------------------------------------------- deterministic final reduce + margin
__global__ void ot_final(const float* __restrict__ rd, const int* __restrict__ label,
                         float* __restrict__ out) {
  __shared__ float red[256];
  const int t = threadIdx.x;
  red[t] = rd[t] + rd[t + 256] + rd[t + 512] + rd[t + 768];
  __syncthreads();
  for (int off = 128; off > 0; off >>= 1) {
    if (t < off) red[t] += red[t + off];
    __syncthreads();
  }
  if (t == 0) {
    float d = red[0];
    out[0] = (label[0] != 0) ? d : fmaxf(0.7f - d, 0.f);
  }
}

extern "C" void kernel_launch(void* const* d_in, const int* in_sizes, int n_in,
                              void* d_out, int out_size, void* d_ws, size_t ws_size,
                              hipStream_t stream) {
  (void)in_sizes; (void)n_in; (void)out_size; (void)ws_size;
  const float* x   = (const float*)d_in[0];
  const float* att = (const float*)d_in[1];
  const int* label = (const int*)d_in[2];
  float* ws  = (float*)d_ws;
  float* out = (float*)d_out;

  ot_prep<<<36, 256, 0, stream>>>(x, att, ws);
  ot_gemm<<<1024, 256, 0, stream>>>(x, ws);
  for (int it = 0; it < OT_ITERS; ++it) {
    ot_u<<<1024, 256, 0, stream>>>(ws + OFF_M, ws + OFF_V, ws + OFF_LOG, ws + OFF_U);
    ot_v<<<128, 256, 0, stream>>>(ws + OFF_M, ws + OFF_U, ws + OFF_LOG, ws + OFF_V);
  }
  ot_pt<<<1024, 256, 0, stream>>>(x, att, ws + OFF_M, ws + OFF_U, ws + OFF_V, ws + OFF_RD);
  ot_final<<<1, 256, 0, stream>>>(ws + OFF_RD, label, out);
}